// HierarchicalEncoder_39075612459516
// MI455X (gfx1250) — compile-verified
//
#include <hip/hip_runtime.h>

// CDNA5 / gfx1250 implementation of the hierarchical GCN encoder.
// f32 WMMA (V_WMMA_F32_16X16X4_F32) for all dense GEMMs; edge aggregation is
// L2-resident (25.6MB activations vs 192MB L2) wave-per-edge atomic scatter.

typedef __attribute__((ext_vector_type(2))) float v2f;
typedef __attribute__((ext_vector_type(8))) float v8f;

#define DIMF 128

// ---------- small utility kernels ----------
__global__ void k_fill(float* p, float v, int n) {
    int i = blockIdx.x * blockDim.x + threadIdx.x;
    if (i < n) p[i] = v;
}

__global__ void k_deg(const int* __restrict__ dst, int E, float* deg) {
    int e = blockIdx.x * blockDim.x + threadIdx.x;
    if (e < E) atomicAdd(&deg[dst[e]], 1.0f);
}

__global__ void k_rsqrt(float* p, int n) {
    int i = blockIdx.x * blockDim.x + threadIdx.x;
    if (i < n) p[i] = rsqrtf(p[i]);
}

__global__ void k_count(const int* __restrict__ batch, int N, float* cnt) {
    int i = blockIdx.x * blockDim.x + threadIdx.x;
    if (i < N) atomicAdd(&cnt[batch[i]], 1.0f);
}

__global__ void k_invcnt(float* c, int n) {
    int i = blockIdx.x * blockDim.x + threadIdx.x;
    if (i < n) c[i] = 1.0f / fmaxf(c[i], 1.0f);
}

// ---------- f32 WMMA GEMM: Y[M,128] = X[M,K] @ Wm[K,128] (+bias, relu) ----------
// One wave32 computes one 16x16 output tile via V_WMMA_F32_16X16X4_F32.
// A frag (16x4 f32): lanes 0-15 row M=lane hold K={0,1}; lanes 16-31 hold K={2,3}.
// B frag (4x16 f32): mirrored K-split across lane halves, col = lane&15.
// C/D (16x16 f32, 8 VGPRs): VGPR r -> row r (lanes 0-15) / row r+8 (lanes 16-31).
__global__ void k_gemm_wmma(const float* __restrict__ X, const float* __restrict__ Wm,
                            const float* __restrict__ bias, float* __restrict__ Y,
                            int M, int K, int relu) {
    const int wave = threadIdx.x >> 5;
    const int lane = threadIdx.x & 31;
    const int tile = blockIdx.x * 4 + wave;
    const int Mtiles = (M + 15) >> 4;
    const int tm = tile >> 3;          // 8 column tiles cover the 128 output dims
    const int tn = tile & 7;
    if (tm >= Mtiles) return;          // wave-uniform exit keeps EXEC all-ones

    const int half = lane >> 4;        // 0: K=0,1   1: K=2,3
    const int l15  = lane & 15;
    const int row  = tm * 16 + l15;
    const int rowc = row < M ? row : M - 1;   // clamp loads; stores are guarded
    const int col  = tn * 16 + l15;
    const int koff = half * 2;

    const float* xr = X + (long)rowc * K + koff;
    v8f c = {};
#pragma unroll 4
    for (int k4 = 0; k4 < K; k4 += 4) {
        v2f a = *(const v2f*)(xr + k4);               // contiguous pair -> b64 load
        v2f bf;
        bf.x = Wm[(k4 + koff) * DIMF + col];
        bf.y = Wm[(k4 + koff + 1) * DIMF + col];
        c = __builtin_amdgcn_wmma_f32_16x16x4_f32(
                /*neg_a=*/false, a, /*neg_b=*/false, bf,
                /*c_mod=*/(short)0, c, /*reuse_a=*/false, /*reuse_b=*/false);
    }
#pragma unroll
    for (int r = 0; r < 8; ++r) {
        int rr = tm * 16 + r + half * 8;
        if (rr < M) {
            float v = c[r];
            if (bias) v += bias[col];
            if (relu) v = fmaxf(v, 0.0f);
            Y[(long)rr * DIMF + col] = v;
        }
    }
}

// ---------- aggregation ----------
// h[i,:] = tmp[i,:] * dinv[i]^2 + b[:]   (self-loop term + bias)
__global__ void k_agg_init(const float* __restrict__ tmp, const float* __restrict__ dinv,
                           const float* __restrict__ bl, float* __restrict__ h, long n) {
    long i = (long)blockIdx.x * blockDim.x + threadIdx.x;
    if (i < n) {
        int r = (int)(i >> 7), d = (int)(i & 127);
        float di = dinv[r];
        h[i] = tmp[i] * di * di + bl[d];
    }
}

// one wave per edge: gather float4 row chunk, scale, atomic scatter-add
__global__ void k_agg_edges(const int* __restrict__ ei, int E,
                            const float* __restrict__ tmp, const float* __restrict__ dinv,
                            float* __restrict__ h) {
    int e = (blockIdx.x * blockDim.x + threadIdx.x) >> 5;
    int lane = threadIdx.x & 31;
    if (e >= E) return;
    int s = ei[e], d = ei[E + e];
    float nrm = dinv[s] * dinv[d];
    const float4 v = *(const float4*)(tmp + (long)s * DIMF + lane * 4);
    float* o = h + (long)d * DIMF + lane * 4;
    atomicAdd(o + 0, v.x * nrm);
    atomicAdd(o + 1, v.y * nrm);
    atomicAdd(o + 2, v.z * nrm);
    atomicAdd(o + 3, v.w * nrm);
}

// ---------- batchnorm ----------
// block = 128 threads (one per feature dim), each block reduces 256 rows
__global__ void k_bn_stats(const float* __restrict__ h, int N, float* sums) {
    int d = threadIdx.x;
    int r0 = blockIdx.x * 256;
    int r1 = min(r0 + 256, N);
    float s = 0.0f, s2 = 0.0f;
    for (int r = r0; r < r1; ++r) {
        float v = h[(long)r * DIMF + d];
        s += v; s2 += v * v;
    }
    atomicAdd(&sums[d], s);
    atomicAdd(&sums[DIMF + d], s2);
}

__global__ void k_bn_finalize(const float* __restrict__ sums, const float* __restrict__ g,
                              const float* __restrict__ be, float* ss, float invN) {
    int d = threadIdx.x;
    float mu  = sums[d] * invN;
    float var = sums[DIMF + d] * invN - mu * mu;      // biased variance
    float sc  = g[d] * rsqrtf(var + 1e-5f);
    ss[d] = sc;
    ss[DIMF + d] = be[d] - mu * sc;
}

__global__ void k_bn_apply(float* h, const float* __restrict__ ss, long n) {
    long i = (long)blockIdx.x * blockDim.x + threadIdx.x;
    if (i < n) {
        int d = (int)(i & 127);
        h[i] = fmaxf(h[i] * ss[d] + ss[DIMF + d], 0.0f);
    }
}

// ---------- per-graph mean pooling into fused representation ----------
__global__ void k_pool(const float* __restrict__ h, const int* __restrict__ batch,
                       const float* __restrict__ invc, float* reps, int loff, int ldr) {
    int i = blockIdx.x;       // node
    int d = threadIdx.x;      // dim (128)
    int g = batch[i];
    atomicAdd(&reps[(long)g * ldr + loff + d], h[(long)i * DIMF + d] * invc[g]);
}

// ---------- host ----------
extern "C" void kernel_launch(void* const* d_in, const int* in_sizes, int n_in,
                              void* d_out, int out_size, void* d_ws, size_t ws_size,
                              hipStream_t stream) {
    const float* x     = (const float*)d_in[0];
    const int*   ei    = (const int*)d_in[1];
    const int*   batch = (const int*)d_in[2];
    const float* W     = (const float*)d_in[3];
    const float* b     = (const float*)d_in[4];
    const float* gamma = (const float*)d_in[5];
    const float* beta  = (const float*)d_in[6];
    const float* Wf    = (const float*)d_in[7];
    const float* bf    = (const float*)d_in[8];
    float* out = (float*)d_out;

    const int N  = in_sizes[2];                    // 50000
    const int E  = in_sizes[1] / 2;                // 800000
    const int nlayers = in_sizes[3] / (DIMF * DIMF); // 6
    const int G  = out_size / DIMF;                // 512
    const int Kf = in_sizes[7] / DIMF;             // 384
    const int nlvl = Kf / DIMF;                    // 3
    const int per  = nlayers / nlvl;               // 2

    // workspace carve-out (floats)
    float* ws   = (float*)d_ws;
    float* tmp  = ws;                              // N*128  (h @ W result)
    float* h    = tmp + (size_t)N * DIMF;          // N*128  (activations)
    float* dinv = h + (size_t)N * DIMF;            // N
    float* sums = dinv + (size_t)((N + 63) & ~63); // 256
    float* ss   = sums + 256;                      // 256
    float* invc = ss + 256;                        // G
    float* reps = invc + (size_t)((G + 63) & ~63); // G * Kf

    const long ND = (long)N * DIMF;
    const int T = 256;

    // degree (self-loop => init 1.0) -> dinv = rsqrt(deg)
    k_fill<<<(N + T - 1) / T, T, 0, stream>>>(dinv, 1.0f, N);
    k_deg<<<(E + T - 1) / T, T, 0, stream>>>(ei + E, E, dinv);
    k_rsqrt<<<(N + T - 1) / T, T, 0, stream>>>(dinv, N);

    // per-graph counts -> 1/max(c,1)
    k_fill<<<(G + T - 1) / T, T, 0, stream>>>(invc, 0.0f, G);
    k_count<<<(N + T - 1) / T, T, 0, stream>>>(batch, N, invc);
    k_invcnt<<<(G + T - 1) / T, T, 0, stream>>>(invc, G);

    // zero fused representation buffer
    k_fill<<<(G * Kf + T - 1) / T, T, 0, stream>>>(reps, 0.0f, G * Kf);

    const int Mtiles = (N + 15) >> 4;
    const int gemmBlocks = (Mtiles * 8 + 3) / 4;     // 4 waves (tiles) per block

    const float* cur = x;
    int li = 0;
    for (int lvl = 0; lvl < nlvl; ++lvl) {
        for (int j = 0; j < per; ++j, ++li) {
            // tmp = cur @ W[li]
            k_gemm_wmma<<<gemmBlocks, 128, 0, stream>>>(
                cur, W + (size_t)li * DIMF * DIMF, nullptr, tmp, N, DIMF, 0);
            // h = self-loop + bias, then scatter-add edges
            k_agg_init<<<(int)((ND + T - 1) / T), T, 0, stream>>>(
                tmp, dinv, b + (size_t)li * DIMF, h, ND);
            k_agg_edges<<<(E * 32 + T - 1) / T, T, 0, stream>>>(ei, E, tmp, dinv, h);
            // batchnorm + relu (in place on h)
            k_fill<<<1, 256, 0, stream>>>(sums, 0.0f, 256);
            k_bn_stats<<<(N + 255) / 256, DIMF, 0, stream>>>(h, N, sums);
            k_bn_finalize<<<1, DIMF, 0, stream>>>(
                sums, gamma + (size_t)li * DIMF, beta + (size_t)li * DIMF, ss, 1.0f / (float)N);
            k_bn_apply<<<(int)((ND + T - 1) / T), T, 0, stream>>>(h, ss, ND);
            cur = h;
        }
        // mean-pool this level into fused reps
        k_pool<<<N, DIMF, 0, stream>>>(h, batch, invc, reps, lvl * DIMF, Kf);
    }

    // out = relu(reps(G x Kf) @ Wf + bf)
    const int MtilesF = (G + 15) >> 4;
    k_gemm_wmma<<<(MtilesF * 8 + 3) / 4, 128, 0, stream>>>(reps, Wf, bf, out, G, Kf, 1);
}